// connect_loss_9251359555892
// MI455X (gfx1250) — compile-verified
//
#include <hip/hip_runtime.h>
#include <math.h>
#include <stdint.h>

typedef float v2f __attribute__((ext_vector_type(2)));
typedef float v8f __attribute__((ext_vector_type(8)));
typedef unsigned int v4u __attribute__((ext_vector_type(4)));
typedef int v8i __attribute__((ext_vector_type(8)));
typedef int v4i __attribute__((ext_vector_type(4)));

#define B_ 8
#define H_ 512
#define W_ 512
#define HW_ (H_ * W_)

// fast sigmoid: v_exp_f32 + v_rcp_f32 (avoids IEEE v_div_* chains)
__device__ __forceinline__ float sigmoidf(float x) {
    return __builtin_amdgcn_rcpf(1.0f + __expf(-x));
}

__device__ __forceinline__ float bce_term(float p, float t) {
    float lp = fmaxf(__logf(p), -100.0f);
    float ln = fmaxf(__logf(1.0f - p), -100.0f);
    return -(t * lp + (1.0f - t) * ln);
}

__device__ __forceinline__ v8f wmma_f32(v2f a, v2f b, v8f c) {
    // D = A(16x4) * B(4x16) + C(16x16), f32, wave32
    return __builtin_amdgcn_wmma_f32_16x16x4_f32(
        /*neg_a=*/false, a, /*neg_b=*/false, b,
        /*c_mod=*/(short)0, c, /*reuse_a=*/false, /*reuse_b=*/false);
}

// ---------------------------------------------------------------------------
// Kernel 1: conn_l = mean BCE(sigmoid(c_map), con_target). con_target built
// on the fly from target via zero-padded shifts. Per-block partial sums.
// ---------------------------------------------------------------------------
__global__ void conn_bce_kernel(const float* __restrict__ cmap,
                                const float* __restrict__ target,
                                float* __restrict__ partial) {
    const unsigned N = (unsigned)B_ * 8u * HW_;
    float acc = 0.0f;
    for (unsigned e = blockIdx.x * blockDim.x + threadIdx.x; e < N;
         e += gridDim.x * blockDim.x) {
        unsigned w = e % W_;
        unsigned r1 = e / W_;
        unsigned h = r1 % H_;
        unsigned r2 = r1 / H_;
        int c = (int)(r2 % 8u);
        int b = (int)(r2 / 8u);
        int dr, dc;
        if (c < 3)      { dr = -1; dc = c - 1; }
        else if (c < 5) { dr = 0;  dc = (c == 3) ? -1 : 1; }
        else            { dr = 1;  dc = c - 6; }
        float t0 = target[(unsigned)b * HW_ + h * W_ + w];
        int hh = (int)h + dr, ww = (int)w + dc;
        float ts = (hh >= 0 && hh < H_ && ww >= 0 && ww < W_)
                       ? target[(unsigned)b * HW_ + (unsigned)hh * W_ + (unsigned)ww]
                       : 0.0f;
        float con = t0 * ts;
        float p = sigmoidf(cmap[e]);
        acc += bce_term(p, con);
    }
    __shared__ float sh[256];
    sh[threadIdx.x] = acc;
    __syncthreads();
    for (int s = 128; s > 0; s >>= 1) {
        if ((int)threadIdx.x < s) sh[threadIdx.x] += sh[threadIdx.x + s];
        __syncthreads();
    }
    if (threadIdx.x == 0) partial[blockIdx.x] = sh[0];
}

// ---------------------------------------------------------------------------
// Kernel 2: vertical einsums as WMMA f32 GEMMs.
//   j=0: vd(P7)=Vm^T@P7, j=1: vd(P6), j=2: vd(P5),
//   j=3: vu(P2)=Vm@P2,   j=4: vu(P1), j=5: vu(P0)
// P = sigmoid(c_map channel). Each wave: 16(M) x 64(N) output strip, K=512.
// A-element addressing is branchless (stride select) so the K-loop carries
// no divergent EXEC-masked load paths.
// ---------------------------------------------------------------------------
__global__ void vstage_kernel(const float* __restrict__ cmap,
                              const float* __restrict__ Vm,
                              float* __restrict__ Abuf) {
    int wid = blockIdx.x * (blockDim.x >> 5) + (threadIdx.x >> 5);
    int lane = threadIdx.x & 31;
    int lm = lane & 15;
    int hi = lane >> 4;

    int j = wid >> 11;       // 2048 waves per map (8 batches * 32 * 8)
    int r = wid & 2047;
    int b = r >> 8;          // 256 waves per (j,b)
    int t = r & 255;
    int h0 = (t >> 3) << 4;  // 32 row tiles
    int w0 = (t & 7) << 6;   // 8 col groups of 64

    int ch = (j == 0) ? 7 : (j == 1) ? 6 : (j == 2) ? 5 : (j == 3) ? 2 : (j == 4) ? 1 : 0;
    int tr = (j < 3);        // 1 => A(m,k) = Vm[k,m] (Vm^T)
    int sk = tr ? H_ : 1;    // stride along k
    int sm = tr ? 1 : H_;    // stride along m

    const float* X = cmap + ((size_t)b * 8 + ch) * HW_;
    v8f acc[4];
    acc[0] = v8f{}; acc[1] = v8f{}; acc[2] = v8f{}; acc[3] = v8f{};

    int mrow = h0 + lm;
    const float* Vrow = Vm + (size_t)mrow * sm;
    for (int k0 = 0; k0 < H_; k0 += 4) {
        int ka = k0 + hi * 2;
        v2f a;
        a.x = Vrow[(size_t)(ka + 0) * sk];
        a.y = Vrow[(size_t)(ka + 1) * sk];
#pragma unroll
        for (int nt = 0; nt < 4; ++nt) {
            int n = w0 + nt * 16 + lm;
            v2f bb;
            bb.x = sigmoidf(X[(ka + 0) * W_ + n]);
            bb.y = sigmoidf(X[(ka + 1) * W_ + n]);
            acc[nt] = wmma_f32(a, bb, acc[nt]);
        }
    }

    float* Ob = Abuf + ((size_t)j * B_ + b) * HW_;
#pragma unroll
    for (int nt = 0; nt < 4; ++nt) {
#pragma unroll
        for (int rr = 0; rr < 8; ++rr) {
            int row = h0 + rr + hi * 8;
            int col = w0 + nt * 16 + lm;
            Ob[row * W_ + col] = acc[nt][rr];
        }
    }
}

// ---------------------------------------------------------------------------
// Kernel 3: horizontal einsums (WMMA, K=512) + votes + loss partials.
// Block = 4 waves sharing one 16-column strip (same w0, h0..h0+63).
// The Hm^T strip (16 rows x 512) is staged into LDS once per block by the
// Tensor Data Mover (tensor_load_to_lds) with D# padding (2 DWORDs / 256)
// so per-lane row reads are bank-conflict free. Wave0 issues the TDM and
// waits on TENSORcnt, then the block barriers.
// ---------------------------------------------------------------------------
#define HMT_STRIDE 516  // 512 + 2 pad DWORDs per 256 DWORDs

__global__ void vote_loss_kernel(const float* __restrict__ cmap,
                                 const float* __restrict__ target,
                                 const float* __restrict__ Hm,
                                 const float* __restrict__ Abuf,
                                 float* __restrict__ partial) {
    __shared__ float shHmT[16 * HMT_STRIDE];  // ~33 KB

    int lane = threadIdx.x & 31;
    int slot = threadIdx.x >> 5;  // wave within block (0..3)
    int lm = lane & 15;
    int hi = lane >> 4;

    int blk = blockIdx.x;          // 2048 blocks
    int b = blk >> 8;              // 256 blocks per batch
    int rem = blk & 255;
    int hblk = rem >> 5;           // 8 row-blocks of 64
    int w0 = (rem & 31) << 4;      // 32 column strips of 16
    int h0 = (hblk << 6) + (slot << 4);
    int wid = blk * 4 + slot;      // global wave id, b == wid >> 10

#if __has_builtin(__builtin_amdgcn_tensor_load_to_lds)
    if (slot == 0) {
        // D# group0: count=1, lds_addr, global_addr, type=2
        uint64_t ga = (uint64_t)(uintptr_t)(Hm + (size_t)w0 * W_);
        uint32_t lds = (uint32_t)(uintptr_t)&shHmT[0];  // low 32 bits = LDS offset
        v4u g0;
        g0.x = 0x1u;
        g0.y = lds;
        g0.z = (uint32_t)(ga & 0xffffffffu);
        g0.w = (uint32_t)((ga >> 32) & 0x01ffffffu) | (2u << 30);
        // D# group1: data_size=4B(code 2), pad_enable, pad_interval=256dw(7),
        // pad_amount=2dw(code 1); tensor 512x16, tile 512x16, dim0 stride 512
        unsigned td0 = 512, td1 = 16, tl0 = 512, tl1 = 16, st0 = 512;
        v8i g1;
        g1[0] = (int)((2u << 16) | (1u << 20) | (7u << 22) | (1u << 25));
        g1[1] = (int)((td0 & 0xffffu) << 16);
        g1[2] = (int)(((td0 >> 16) & 0xffffu) | ((td1 & 0xffffu) << 16));
        g1[3] = (int)(((td1 >> 16) & 0xffffu) | ((tl0 & 0xffffu) << 16));
        g1[4] = (int)(tl1 & 0xffffu);
        g1[5] = (int)st0;
        g1[6] = 0;
        g1[7] = 0;
        v4i gz4 = v4i{0, 0, 0, 0};
        v8i gz8 = v8i{0, 0, 0, 0, 0, 0, 0, 0};
        // 6-arg form on this toolchain: (g0, g1, g2, g3, extra, cpol)
        __builtin_amdgcn_tensor_load_to_lds(g0, g1, gz4, gz4, gz8, 0);
        __builtin_amdgcn_s_wait_tensorcnt(0);
    }
#else
    for (int i = threadIdx.x; i < 16 * 512; i += blockDim.x) {
        int r = i >> 9, k = i & 511;
        shHmT[r * HMT_STRIDE + k + ((k >> 8) << 1)] = Hm[(w0 + r) * W_ + k];
    }
#endif
    __syncthreads();

    const float* Bf0 = Abuf + ((size_t)0 * B_ + b) * HW_;  // vd(P7)
    const float* Bf1 = Abuf + ((size_t)1 * B_ + b) * HW_;  // vd(P6)
    const float* Bf2 = Abuf + ((size_t)2 * B_ + b) * HW_;  // vd(P5)
    const float* Bf3 = Abuf + ((size_t)3 * B_ + b) * HW_;  // vu(P2)
    const float* Bf4 = Abuf + ((size_t)4 * B_ + b) * HW_;  // vu(P1)
    const float* Bf5 = Abuf + ((size_t)5 * B_ + b) * HW_;  // vu(P0)
    const float* P3 = cmap + ((size_t)b * 8 + 3) * HW_;
    const float* P4 = cmap + ((size_t)b * 8 + 4) * HW_;

    v8f u0 = v8f{}, u2 = v8f{}, u3 = v8f{}, u4 = v8f{}, u5 = v8f{}, u7 = v8f{};
    int mrow = h0 + lm;
    int ncol = w0 + lm;

    for (int k0 = 0; k0 < W_; k0 += 4) {
        int ka = k0 + hi * 2;
        // B fragments: Hm[k, ncol] from global (coalesced), Hm[ncol, k] from LDS
        v2f bn, bt;
        bn.x = Hm[(ka + 0) * W_ + ncol];
        bn.y = Hm[(ka + 1) * W_ + ncol];
        int i0 = ka, i1 = ka + 1;
        bt.x = shHmT[lm * HMT_STRIDE + i0 + ((i0 >> 8) << 1)];
        bt.y = shHmT[lm * HMT_STRIDE + i1 + ((i1 >> 8) << 1)];

        // A fragments: adjacent k-pairs, 8B aligned -> single b64 loads
        int ro = mrow * W_ + ka;
        v2f a0 = *(const v2f*)(Bf0 + ro);
        v2f a2 = *(const v2f*)(Bf2 + ro);
        v2f a5 = *(const v2f*)(Bf3 + ro);
        v2f a7 = *(const v2f*)(Bf5 + ro);
        v2f c4 = *(const v2f*)(P4 + ro);
        v2f c3 = *(const v2f*)(P3 + ro);
        v2f a3, a4;
        a3.x = sigmoidf(c4.x); a3.y = sigmoidf(c4.y);
        a4.x = sigmoidf(c3.x); a4.y = sigmoidf(c3.y);

        u0 = wmma_f32(a0, bn, u0);  // right_bottom = vd(P7) @ Hm
        u2 = wmma_f32(a2, bt, u2);  // left_bottom  = vd(P5) @ Hm^T
        u3 = wmma_f32(a3, bn, u3);  // right        = P4 @ Hm
        u4 = wmma_f32(a4, bt, u4);  // left         = P3 @ Hm^T
        u5 = wmma_f32(a5, bn, u5);  // right_above  = vu(P2) @ Hm
        u7 = wmma_f32(a7, bt, u7);  // left_above   = vu(P0) @ Hm^T
    }

    float bceAcc = 0.0f, eNum = 0.0f, eDen = 0.0f;
    float predS = 0.0f, interS = 0.0f, tS = 0.0f;

#pragma unroll
    for (int rr = 0; rr < 8; ++rr) {
        int h = h0 + rr + hi * 8;
        int w = w0 + lm;
        int off = h * W_ + w;
        size_t cbase = (size_t)b * 8 * HW_ + (size_t)off;

        float p0 = sigmoidf(cmap[cbase + 0 * HW_]);
        float p1 = sigmoidf(cmap[cbase + 1 * HW_]);
        float p2 = sigmoidf(cmap[cbase + 2 * HW_]);
        float p3 = sigmoidf(cmap[cbase + 3 * HW_]);
        float p4 = sigmoidf(cmap[cbase + 4 * HW_]);
        float p5 = sigmoidf(cmap[cbase + 5 * HW_]);
        float p6 = sigmoidf(cmap[cbase + 6 * HW_]);
        float p7 = sigmoidf(cmap[cbase + 7 * HW_]);

        float v0 = p0 * u0[rr];
        float v1 = p1 * Bf1[off];      // bottom = vd(P6)
        float v2 = p2 * u2[rr];
        float v3 = p3 * u3[rr];
        float v4 = p4 * u4[rr];
        float v5 = p5 * u5[rr];
        float v6 = p6 * Bf4[off];      // up = vu(P1)
        float v7 = p7 * u7[rr];

        float pred = fmaxf(fmaxf(fmaxf(v0, v1), fmaxf(v2, v3)),
                           fmaxf(fmaxf(v4, v5), fmaxf(v6, v7)));
        float vmin = fminf(fminf(fminf(v0, v1), fminf(v2, v3)),
                           fminf(fminf(v4, v5), fminf(v6, v7)));

        float tv = target[(size_t)b * HW_ + off];
        bceAcc += bce_term(pred, tv);

        // edge mask: tv==1 and 1 <= (#set 8-neighbors, zero pad) <= 7
        float s = 0.0f;
        if (tv > 0.5f) {
#pragma unroll
            for (int c = 0; c < 8; ++c) {
                int dr, dc;
                if (c < 3)      { dr = -1; dc = c - 1; }
                else if (c < 5) { dr = 0;  dc = (c == 3) ? -1 : 1; }
                else            { dr = 1;  dc = c - 6; }
                int hh = h + dr, ww = w + dc;
                if (hh >= 0 && hh < H_ && ww >= 0 && ww < W_)
                    s += target[(size_t)b * HW_ + hh * W_ + ww];
            }
        }
        float edge = (tv > 0.5f && s > 0.5f && s < 7.5f) ? 1.0f : 0.0f;
        float pmin = vmin * edge;
        eNum += -fmaxf(__logf(1.0f - pmin), -100.0f);
        eDen += pmin;

        predS += pred;
        interS += pred * tv;
        tS += tv;
    }

    // wave32 reduction
    for (int o = 16; o > 0; o >>= 1) {
        bceAcc += __shfl_down(bceAcc, o);
        eNum   += __shfl_down(eNum, o);
        eDen   += __shfl_down(eDen, o);
        predS  += __shfl_down(predS, o);
        interS += __shfl_down(interS, o);
        tS     += __shfl_down(tS, o);
    }
    if (lane == 0) {
        float* p = partial + (size_t)wid * 6;
        p[0] = bceAcc; p[1] = eNum; p[2] = eDen;
        p[3] = predS;  p[4] = interS; p[5] = tS;
    }
}

// ---------------------------------------------------------------------------
// Kernel 4: final deterministic reduction -> scalar loss
// ---------------------------------------------------------------------------
__global__ void finalize_kernel(const float* __restrict__ connPart,
                                const float* __restrict__ tilePart,
                                float* __restrict__ out) {
    __shared__ float sh[256];
    int tid = threadIdx.x;

    float connAcc = 0.0f;
    for (int i = tid; i < 4096; i += 256) connAcc += connPart[i];

    float bce = 0.0f, eN = 0.0f, eD = 0.0f;
    float pS[8] = {}, iS[8] = {}, tS[8] = {};
    for (int i = tid; i < 8192; i += 256) {
        const float* p = tilePart + (size_t)i * 6;
        bce += p[0]; eN += p[1]; eD += p[2];
        int b = i >> 10;
        pS[b] += p[3]; iS[b] += p[4]; tS[b] += p[5];
    }

    auto red = [&](float v) -> float {
        sh[tid] = v;
        __syncthreads();
        for (int s = 128; s > 0; s >>= 1) {
            if (tid < s) sh[tid] += sh[tid + s];
            __syncthreads();
        }
        float r = sh[0];
        __syncthreads();
        return r;
    };

    connAcc = red(connAcc);
    bce = red(bce);
    eN = red(eN);
    eD = red(eD);

    float dice = 0.0f;
    for (int b = 0; b < 8; ++b) {
        float pb = red(pS[b]);
        float ib = red(iS[b]);
        float tb = red(tS[b]);
        dice += 1.0f - (2.0f * ib + 1e-4f) / (tb + pb + 1e-4f);
    }

    if (tid == 0) {
        float conn_l = connAcc / (float)((size_t)B_ * 8 * HW_);
        float bce_l = bce / (float)((size_t)B_ * HW_);
        float edge_l = eN / eD;
        float dice_l = dice / (float)B_;
        out[0] = bce_l + conn_l + edge_l + dice_l;
    }
}

extern "C" void kernel_launch(void* const* d_in, const int* in_sizes, int n_in,
                              void* d_out, int out_size, void* d_ws, size_t ws_size,
                              hipStream_t stream) {
    const float* cmap = (const float*)d_in[0];   // (8,8,512,512)
    const float* target = (const float*)d_in[1]; // (8,1,512,512)
    const float* Hm = (const float*)d_in[2];     // (512,512)
    const float* Vm = (const float*)d_in[3];     // (512,512)
    float* out = (float*)d_out;

    float* ws = (float*)d_ws;
    float* Abuf = ws;                                  // 6 * B * H * W floats
    float* tilePart = ws + (size_t)6 * B_ * HW_;       // 8192 * 6 floats
    float* connPart = tilePart + (size_t)8192 * 6;     // 4096 floats

    conn_bce_kernel<<<4096, 256, 0, stream>>>(cmap, target, connPart);
    vstage_kernel<<<3072, 128, 0, stream>>>(cmap, Vm, Abuf);
    vote_loss_kernel<<<2048, 128, 0, stream>>>(cmap, target, Hm, Abuf, tilePart);
    finalize_kernel<<<1, 256, 0, stream>>>(connPart, tilePart, out);
}